// SpatialEncoder_38122129719384
// MI455X (gfx1250) — compile-verified
//
#include <hip/hip_runtime.h>

typedef float v2f __attribute__((ext_vector_type(2)));
typedef float v8f __attribute__((ext_vector_type(8)));

#define NEG_SLOPE 0.2f

// ---------------------------------------------------------------------------
// fp32 WMMA GEMM: Y[N,D] = X[N,F] @ W[F,D].  D == 64.
// Block = 128 threads (4 waves). Wave w computes the 16x16 tile at
// (m0 = 16*blockIdx.x, n0 = 16*w), looping K in steps of 4 with
// V_WMMA_F32_16X16X4_F32.
// A 16x4 layout : lane L -> M = L&15, VGPR0 = K = khalf, VGPR1 = K = khalf+1,
//                 khalf = 2*(L>>4)
// B 4x16 layout : lane L -> N = L&15, VGPR0 = K = khalf, VGPR1 = K = khalf+1
// C/D 16x16     : VGPR r -> M = r + 8*(L>>4), N = L&15
// ---------------------------------------------------------------------------
__global__ void gemm_wmma_f32(const float* __restrict__ X,
                              const float* __restrict__ W,
                              float* __restrict__ Y,
                              int N, int F, int D) {
  const int wave = threadIdx.x >> 5;
  const int lane = threadIdx.x & 31;
  const int m0 = blockIdx.x * 16;
  const int n0 = wave * 16;
  if (n0 >= D) return;

  const int row   = lane & 15;
  const int khalf = (lane >> 4) << 1;          // 0 or 2
  int mrow = m0 + row;
  if (mrow >= N) mrow = N - 1;                 // clamp loads (stores guarded)
  const int col = n0 + row;

  const float* xp = X + (size_t)mrow * F + khalf;
  v8f acc = {};

  for (int k0 = 0; k0 < F; k0 += 4) {
    v2f a, b;
    a.x = xp[k0];
    a.y = xp[k0 + 1];
    const float* wp = W + (size_t)(k0 + khalf) * D + col;
    b.x = wp[0];
    b.y = wp[D];
    acc = __builtin_amdgcn_wmma_f32_16x16x4_f32(
        /*neg_a=*/false, a, /*neg_b=*/false, b,
        /*c_mod=*/(short)0, acc, /*reuse_a=*/false, /*reuse_b=*/false);
  }

  const int mbase = m0 + ((lane >> 4) << 3);
#pragma unroll
  for (int r = 0; r < 8; ++r) {
    const int mr = mbase + r;
    if (mr < N) Y[(size_t)mr * D + col] = acc[r];
  }
}

// ---------------------------------------------------------------------------
// Per-call buffer init: acc = 0, m = -inf, s = 0
// ---------------------------------------------------------------------------
__global__ void init_buffers(float* __restrict__ mmax, float* __restrict__ ssum,
                             float* __restrict__ acc, int N) {
  const int t = blockIdx.x * blockDim.x + threadIdx.x;
  if (t < N * 64) acc[t] = 0.0f;
  if (t < N * 4) {
    mmax[t] = -__builtin_inff();
    ssum[t] = 0.0f;
  }
}

// float atomic-max via monotone integer punning (init value must be -inf)
__device__ __forceinline__ void atomicMaxFloat(float* addr, float val) {
  if (val >= 0.0f)
    atomicMax((int*)addr, __float_as_int(val));
  else
    atomicMin((unsigned int*)addr, __float_as_uint(val));
}

// ---------------------------------------------------------------------------
// Edge pass 1: one thread per (edge, head).
// logits[e,h] = att[h,:] . LeakyReLU(xl[src,h,:] + xr[dst,h,:])
// mmax[dst,h] = max(mmax[dst,h], logits[e,h])
// Edges e < E come from edge_index; e in [E, E+N) are self-loops.
// ---------------------------------------------------------------------------
__global__ void edge_logits_kernel(const int* __restrict__ ei, int E, int N,
                                   const float* __restrict__ xl,
                                   const float* __restrict__ xr,
                                   const float* __restrict__ att,
                                   float* __restrict__ logits,
                                   float* __restrict__ mmax) {
  const int t = blockIdx.x * blockDim.x + threadIdx.x;
  const int total = (E + N) * 4;
  if (t >= total) return;
  const int h = t & 3;
  const int e = t >> 2;
  int src, dst;
  if (e < E) { src = ei[e]; dst = ei[E + e]; }
  else       { src = e - E; dst = src; }

  const float4* pl = (const float4*)(xl + (size_t)src * 64 + h * 16);
  const float4* pr = (const float4*)(xr + (size_t)dst * 64 + h * 16);
  const float4* pa = (const float4*)(att + h * 16);

  float s = 0.0f;
#pragma unroll
  for (int q = 0; q < 4; ++q) {
    const float4 a4 = pa[q];
    const float4 l4 = pl[q];
    const float4 r4 = pr[q];
    float v;
    v = l4.x + r4.x; v = (v > 0.0f) ? v : NEG_SLOPE * v; s += v * a4.x;
    v = l4.y + r4.y; v = (v > 0.0f) ? v : NEG_SLOPE * v; s += v * a4.y;
    v = l4.z + r4.z; v = (v > 0.0f) ? v : NEG_SLOPE * v; s += v * a4.z;
    v = l4.w + r4.w; v = (v > 0.0f) ? v : NEG_SLOPE * v; s += v * a4.w;
  }
  logits[t] = s;
  atomicMaxFloat(mmax + (size_t)dst * 4 + h, s);
}

// ---------------------------------------------------------------------------
// Edge pass 2 (fused softmax-denominator + unnormalized message accumulate):
// p = exp(logit - m[dst]); s[dst,h] += p; acc[dst,h,:] += xl[src,h,:] * p
// Normalization happens in the node-level finalize (acc / s).
// ---------------------------------------------------------------------------
__global__ void edge_accum_kernel(const int* __restrict__ ei, int E, int N,
                                  const float* __restrict__ xl,
                                  const float* __restrict__ logits,
                                  const float* __restrict__ mmax,
                                  float* __restrict__ ssum,
                                  float* __restrict__ acc) {
  const int t = blockIdx.x * blockDim.x + threadIdx.x;
  const int total = (E + N) * 4;
  if (t >= total) return;
  const int h = t & 3;
  const int e = t >> 2;
  int src, dst;
  if (e < E) { src = ei[e]; dst = ei[E + e]; }
  else       { src = e - E; dst = src; }

  const float p = __expf(logits[t] - mmax[(size_t)dst * 4 + h]);
  atomicAdd(ssum + (size_t)dst * 4 + h, p);

  const float* pl = xl + (size_t)src * 64 + h * 16;
  float* po = acc + (size_t)dst * 64 + h * 16;
#pragma unroll
  for (int c = 0; c < 16; ++c) atomicAdd(po + c, pl[c] * p);
}

// ---------------------------------------------------------------------------
// Node finalize: out[i,h,c] = relu(acc[i,h,c] / s[i,h] + bias[h*16+c])
// (self-loop guarantees s > 0 for every node)
// ---------------------------------------------------------------------------
__global__ void finalize_kernel(const float* __restrict__ acc,
                                const float* __restrict__ ssum,
                                const float* __restrict__ bias,
                                float* __restrict__ out, int N) {
  const int t = blockIdx.x * blockDim.x + threadIdx.x;
  if (t >= N * 64) return;
  const int d = t & 63;
  const int node = t >> 6;
  const int h = d >> 4;
  float v = acc[t] / ssum[(size_t)node * 4 + h] + bias[d];
  out[t] = (v > 0.0f) ? v : 0.0f;
}

// ---------------------------------------------------------------------------
// Host-side orchestration (graph-capturable: only kernel launches on stream)
// ---------------------------------------------------------------------------
extern "C" void kernel_launch(void* const* d_in, const int* in_sizes, int n_in,
                              void* d_out, int out_size, void* d_ws, size_t ws_size,
                              hipStream_t stream) {
  const float* x    = (const float*)d_in[0];
  const int*   ei   = (const int*)d_in[1];
  const float* W1l  = (const float*)d_in[2];
  const float* W1r  = (const float*)d_in[3];
  const float* att1 = (const float*)d_in[4];
  const float* b1   = (const float*)d_in[5];
  const float* W2l  = (const float*)d_in[6];
  const float* W2r  = (const float*)d_in[7];
  const float* att2 = (const float*)d_in[8];
  const float* b2   = (const float*)d_in[9];

  const int D = in_sizes[5];           // 64  (= H*C, H=4, C=16)
  const int F = in_sizes[2] / D;       // 128
  const int N = in_sizes[0] / F;       // 100000
  const int E = in_sizes[1] / 2;       // 1600000

  float* out = (float*)d_out;

  // Workspace layout
  float* xl     = (float*)d_ws;                 // N*D
  float* xr     = xl + (size_t)N * D;           // N*D
  float* acc    = xr + (size_t)N * D;           // N*D (layer1 h1 in place, reused layer2)
  float* mmax   = acc + (size_t)N * D;          // N*4
  float* ssum   = mmax + (size_t)N * 4;         // N*4
  float* logits = ssum + (size_t)N * 4;         // (E+N)*4

  const int ET = E + N;
  const int gemmGrid = (N + 15) / 16;
  const int edgeGrid = (ET * 4 + 255) / 256;
  const int nodeGrid = (N * 64 + 255) / 256;

  // ---- Layer 1 ----
  gemm_wmma_f32<<<gemmGrid, 128, 0, stream>>>(x, W1l, xl, N, F, D);
  gemm_wmma_f32<<<gemmGrid, 128, 0, stream>>>(x, W1r, xr, N, F, D);
  init_buffers<<<nodeGrid, 256, 0, stream>>>(mmax, ssum, acc, N);
  edge_logits_kernel<<<edgeGrid, 256, 0, stream>>>(ei, E, N, xl, xr, att1, logits, mmax);
  edge_accum_kernel<<<edgeGrid, 256, 0, stream>>>(ei, E, N, xl, logits, mmax, ssum, acc);
  finalize_kernel<<<nodeGrid, 256, 0, stream>>>(acc, ssum, b1, acc, N);  // in-place -> h1

  // ---- Layer 2 (input = acc holding h1; consumed by the two GEMMs) ----
  gemm_wmma_f32<<<gemmGrid, 128, 0, stream>>>(acc, W2l, xl, N, D, D);
  gemm_wmma_f32<<<gemmGrid, 128, 0, stream>>>(acc, W2r, xr, N, D, D);
  init_buffers<<<nodeGrid, 256, 0, stream>>>(mmax, ssum, acc, N);
  edge_logits_kernel<<<edgeGrid, 256, 0, stream>>>(ei, E, N, xl, xr, att2, logits, mmax);
  edge_accum_kernel<<<edgeGrid, 256, 0, stream>>>(ei, E, N, xl, logits, mmax, ssum, acc);
  finalize_kernel<<<nodeGrid, 256, 0, stream>>>(acc, ssum, b2, out, N);
}